// TopKPooling_10660108829530
// MI455X (gfx1250) — compile-verified
//
#include <hip/hip_runtime.h>
#include <hip/hip_bf16.h>
#include <math.h>

typedef __attribute__((ext_vector_type(2))) float v2f;
typedef __attribute__((ext_vector_type(8))) float v8f;

static constexpr int B  = 8;
static constexpr int N  = 4096;
static constexpr int F  = 256;
static constexpr int K  = 1024;
static constexpr float EPS = 1e-7f;

// ---------------------------------------------------------------------------
// 1) ||p||_2  (single block, 256 threads)
// ---------------------------------------------------------------------------
__global__ void pnorm_kernel(const float* __restrict__ p, float* __restrict__ norm) {
    __shared__ float red[256];
    float v = p[threadIdx.x];
    red[threadIdx.x] = v * v;
    __syncthreads();
    for (int s = 128; s > 0; s >>= 1) {
        if (threadIdx.x < (unsigned)s) red[threadIdx.x] += red[threadIdx.x + s];
        __syncthreads();
    }
    if (threadIdx.x == 0) norm[0] = sqrtf(red[0]);
}

// ---------------------------------------------------------------------------
// 2) scores y[b,n] = dot(x[b,n,:], p) / (||p|| + eps)
//    One wave32 computes 16 rows via V_WMMA_F32_16X16X4_F32, accumulating
//    over K-dim in steps of 4.  B-matrix replicates the p-segment across all
//    16 columns, so D[:,0] holds the 16 dot products.
//
//    ISA layouts (CDNA5 ISA):
//      A 16x4 f32 : lane<16 -> {K=0,K=1}, lane>=16 -> {K=2,K=3}, M = lane&15
//      B 4x16 f32 : lane<16 -> rows K=0/K=1 (N=lane), lane>=16 -> K=2/K=3
//      D 16x16 f32: vgpr v, lane<16 -> (M=v,    N=lane),
//                            lane>=16 -> (M=v+8, N=lane-16)
// ---------------------------------------------------------------------------
__global__ void scores_kernel(const float* __restrict__ x,
                              const float* __restrict__ p,
                              const float* __restrict__ norm,
                              float* __restrict__ y,
                              int rows) {
    const int lane = threadIdx.x & 31;
    const int wave = (blockIdx.x * blockDim.x + threadIdx.x) >> 5;
    const int r0   = wave * 16;
    if (r0 >= rows) return;                 // wave-uniform: EXEC stays all-1s

    const int half = lane >> 4;             // 0: K=0,1   1: K=2,3
    const int l    = lane & 15;             // M (for A) / N (for B)

#if __has_builtin(__builtin_amdgcn_wmma_f32_16x16x4_f32)
    const float* xrow = x + (size_t)(r0 + l) * F;
    v8f c = {};
    #pragma unroll 8
    for (int k0 = 0; k0 < F; k0 += 4) {
        v2f a, bfrag;
        a.x     = xrow[k0 + 2 * half + 0];
        a.y     = xrow[k0 + 2 * half + 1];
        bfrag.x = p[k0 + 2 * half + 0];     // replicate across all N columns
        bfrag.y = p[k0 + 2 * half + 1];
        c = __builtin_amdgcn_wmma_f32_16x16x4_f32(
                false, a, false, bfrag, (short)0, c, false, false);
    }
    const float s = 1.0f / (norm[0] + EPS);
    if (l == 0) {                           // lanes 0 and 16 hold column N=0
        #pragma unroll
        for (int v = 0; v < 8; ++v)
            y[r0 + half * 8 + v] = c[v] * s;
    }
#else
    // Scalar fallback: lanes 0..15 each compute one full dot product.
    if (half == 0) {
        const float* xrow = x + (size_t)(r0 + l) * F;
        float acc = 0.0f;
        #pragma unroll 8
        for (int k = 0; k < F; ++k) acc = fmaf(xrow[k], p[k], acc);
        y[r0 + l] = acc / (norm[0] + EPS);
    }
#endif
}

// ---------------------------------------------------------------------------
// 3) stable top-k by exhaustive ranking (matches jax.lax.top_k ordering:
//    descending value, ties broken by lower index).
//    grid = (N/256, B); y[b,:] staged in 16KB LDS, broadcast reads.
// ---------------------------------------------------------------------------
__global__ void topk_kernel(const float* __restrict__ y, int* __restrict__ idx) {
    __shared__ float sy[N];
    const int b = blockIdx.y;
    for (int i = threadIdx.x; i < N; i += blockDim.x) sy[i] = y[b * N + i];
    __syncthreads();

    const int n = blockIdx.x * blockDim.x + threadIdx.x;
    const float v = sy[n];
    int rank = 0;
    #pragma unroll 8
    for (int m = 0; m < N; ++m) {
        const float w = sy[m];
        rank += (w > v) || (w == v && m < n);
    }
    if (rank < K) idx[b * K + rank] = n;
}

// ---------------------------------------------------------------------------
// 4) A_out[b,i,j] = A[b, idx[b,i], idx[b,j]]
//    grid = (K, B); idx row cached in LDS; coalesced writes, gathered reads.
// ---------------------------------------------------------------------------
__global__ void gatherA_kernel(const float* __restrict__ A,
                               const int* __restrict__ idx,
                               float* __restrict__ out) {
    __shared__ int sidx[K];
    const int b = blockIdx.y;
    const int i = blockIdx.x;
    const int* idxb = idx + b * K;
    for (int t = threadIdx.x; t < K; t += blockDim.x) sidx[t] = idxb[t];
    __syncthreads();

    const int ri = sidx[i];
    const float* Arow = A + ((size_t)b * N + ri) * N;
    float* orow = out + ((size_t)b * K + i) * K;
    #pragma unroll
    for (int j = threadIdx.x; j < K; j += blockDim.x)
        orow[j] = Arow[sidx[j]];
}

// ---------------------------------------------------------------------------
// 5) x_out[b,i,f] = x[b, idx[b,i], f] * tanh(y[b, idx[b,i]])
//    grid = (K, B), block = F (256): fully coalesced.
// ---------------------------------------------------------------------------
__global__ void gatherX_kernel(const float* __restrict__ x,
                               const float* __restrict__ y,
                               const int* __restrict__ idx,
                               float* __restrict__ out) {
    const int b = blockIdx.y;
    const int i = blockIdx.x;
    const int f = threadIdx.x;
    const int n = idx[b * K + i];
    const float g = tanhf(y[b * N + n]);
    out[((size_t)b * K + i) * F + f] = x[((size_t)b * N + n) * F + f] * g;
}

// ---------------------------------------------------------------------------
extern "C" void kernel_launch(void* const* d_in, const int* in_sizes, int n_in,
                              void* d_out, int out_size, void* d_ws, size_t ws_size,
                              hipStream_t stream) {
    const float* A = (const float*)d_in[0];   // [B, N, N]
    const float* x = (const float*)d_in[1];   // [B, N, F]
    const float* p = (const float*)d_in[2];   // [F, 1]

    // workspace: [0..15] norm (padded), then y (B*N f32), then idx (B*K i32)
    float* ws_norm = (float*)d_ws;
    float* ws_y    = ws_norm + 16;
    int*   ws_idx  = (int*)(ws_y + (size_t)B * N);

    float* outA = (float*)d_out;                       // [B, K, K]
    float* outX = outA + (size_t)B * K * K;            // [B, K, F]

    const int rows = B * N;                            // 32768

    pnorm_kernel<<<1, 256, 0, stream>>>(p, ws_norm);

    // 8 waves/block -> 128 rows/block -> 256 blocks
    scores_kernel<<<rows / 128, 256, 0, stream>>>(x, p, ws_norm, ws_y, rows);

    topk_kernel<<<dim3(N / 256, B), 256, 0, stream>>>(ws_y, ws_idx);

    gatherA_kernel<<<dim3(K, B), 256, 0, stream>>>(A, ws_idx, outA);

    gatherX_kernel<<<dim3(K, B), 256, 0, stream>>>(x, ws_y, ws_idx, outX);
}